// VolumetricSampler_78589311582767
// MI455X (gfx1250) — compile-verified
//
#include <hip/hip_runtime.h>
#include <hip/hip_bf16.h>
#include <stdint.h>

#define N_RAYS    16384
#define MAX_STEPS 512
#define GRID_R    128
#define NEAR_T    0.05f
#define FAR_T     3.0f
#define STEP_T    ((FAR_T - NEAR_T) / (float)MAX_STEPS)
#define SAMPLES   (N_RAYS * MAX_STEPS)   /* 8,388,608 */

// d_out layout (float elements), outputs concatenated in reference return order:
static constexpr size_t ORIG_OFF = 0;                          // (SAMPLES,3)
static constexpr size_t DIRS_OFF = (size_t)3 * SAMPLES;        // (SAMPLES,3)
static constexpr size_t S_OFF    = (size_t)6 * SAMPLES;        // (SAMPLES,)
static constexpr size_t E_OFF    = (size_t)7 * SAMPLES;        // (SAMPLES,)
static constexpr size_t PACK_OFF = (size_t)8 * SAMPLES;        // (N_RAYS,2)
static constexpr size_t RIDX_OFF = PACK_OFF + 2 * N_RAYS;      // (SAMPLES,)
static constexpr size_t MASK_OFF = RIDX_OFF + SAMPLES;         // (SAMPLES,)

__device__ __forceinline__ float safe_dir(float d) {
    return (fabsf(d) > 1e-10f) ? d : 1e-10f;
}

// Streaming store: outputs are written once and never re-read by anyone this
// call -> TH_STORE_NT keeps the 8MB occ grid + ray data resident in the 192MB L2
// instead of letting 336MB of output sweep it.
__device__ __forceinline__ void nt_store(float v, float* p) {
    __builtin_nontemporal_store(v, p);
}

__global__ __launch_bounds__(MAX_STEPS)
void vol_sample_kernel(const float* __restrict__ rays_o,
                       const float* __restrict__ rays_d,
                       const float* __restrict__ aabb,
                       const int*   __restrict__ occ,
                       float*       __restrict__ out,
                       int*         __restrict__ counts)
{
    __shared__ float sray[6];

    const int ray = blockIdx.x;
    const int tid = threadIdx.x;

    // ---- CDNA5 async global->LDS stage of this block's ray (o.xyz, d.xyz) ----
    // Flat shared pointers' low 32 bits are the wave-relative LDS byte offset
    // (ISA 10.2 aperture truncation), which is exactly what the async op's
    // LDS-address VGPR wants.
    if (tid < 6) {
        const float* src = (tid < 3) ? (rays_o + 3 * ray + tid)
                                     : (rays_d + 3 * ray + (tid - 3));
        unsigned           lds_addr = (unsigned)(size_t)&sray[tid];
        unsigned long long gaddr    = (unsigned long long)(size_t)src;
        asm volatile("global_load_async_to_lds_b32 %0, %1, off"
                     :: "v"(lds_addr), "v"(gaddr) : "memory");
    }
    asm volatile("s_wait_asynccnt 0" ::: "memory");
    __syncthreads();

    const float ox = sray[0], oy = sray[1], oz = sray[2];
    const float dx = sray[3], dy = sray[4], dz = sray[5];

    const float lo0 = aabb[0], lo1 = aabb[1], lo2 = aabb[2];
    const float hi0 = aabb[3], hi1 = aabb[4], hi2 = aabb[5];

    // Slab test (redundant per thread; ~20 flops, cheaper than any broadcast)
    const float ivx = 1.0f / safe_dir(dx);
    const float ivy = 1.0f / safe_dir(dy);
    const float ivz = 1.0f / safe_dir(dz);
    const float t0x = (lo0 - ox) * ivx, t1x = (hi0 - ox) * ivx;
    const float t0y = (lo1 - oy) * ivy, t1y = (hi1 - oy) * ivy;
    const float t0z = (lo2 - oz) * ivz, t1z = (hi2 - oz) * ivz;
    const float tmin = fmaxf(fmaxf(fminf(t0x, t1x),
                       fmaxf(fminf(t0y, t1y), fminf(t0z, t1z))), NEAR_T);
    const float tmax = fminf(fminf(fmaxf(t0x, t1x),
                       fminf(fmaxf(t0y, t1y), fmaxf(t0z, t1z))), FAR_T);

    // Step interval (match reference rounding: edges = NEAR + k*STEP)
    const float start = NEAR_T + (float)tid * STEP_T;
    const float end   = NEAR_T + (float)(tid + 1) * STEP_T;
    const float mid   = 0.5f * (start + end);

    const float px = ox + dx * mid;
    const float py = oy + dy * mid;
    const float pz = oz + dz * mid;
    const float sx = (px - lo0) / (hi0 - lo0);
    const float sy = (py - lo1) / (hi1 - lo1);
    const float sz = (pz - lo2) / (hi2 - lo2);

    int ix = (int)(sx * (float)GRID_R);  ix = min(max(ix, 0), GRID_R - 1);
    int iy = (int)(sy * (float)GRID_R);  iy = min(max(iy, 0), GRID_R - 1);
    int iz = (int)(sz * (float)GRID_R);  iz = min(max(iz, 0), GRID_R - 1);

    const bool inside = (sx >= 0.0f) & (sx < 1.0f) &
                        (sy >= 0.0f) & (sy < 1.0f) &
                        (sz >= 0.0f) & (sz < 1.0f);
    const bool trange = (start >= tmin) & (end <= tmax);

    bool valid = false;
    if (trange & inside) {
        const int cell = (ix * GRID_R + iy) * GRID_R + iz;
        __builtin_prefetch(&occ[cell], 0, 3);   // gfx1250 global_prefetch_b8 (keep hot)
        valid = (occ[cell] != 0);               // regular (RT) load: stays in L2
    }

    // Per-ray sample count across the 16 wave32 waves of this block
    const int cnt = __syncthreads_count(valid ? 1 : 0);
    if (tid == 0) counts[ray] = cnt;

    const size_t lin  = (size_t)ray * MAX_STEPS + (size_t)tid;
    const float  m    = valid ? 1.0f : 0.0f;
    const float  rayf = (float)ray;              // uniform per block

    nt_store(ox * m, &out[ORIG_OFF + 3 * lin + 0]);
    nt_store(oy * m, &out[ORIG_OFF + 3 * lin + 1]);
    nt_store(oz * m, &out[ORIG_OFF + 3 * lin + 2]);
    nt_store(dx * m, &out[DIRS_OFF + 3 * lin + 0]);
    nt_store(dy * m, &out[DIRS_OFF + 3 * lin + 1]);
    nt_store(dz * m, &out[DIRS_OFF + 3 * lin + 2]);
    nt_store(valid ? start : 0.0f, &out[S_OFF + lin]);
    nt_store(valid ? end   : 0.0f, &out[E_OFF + lin]);
    nt_store(rayf, &out[RIDX_OFF + lin]);        // ray_indices is NOT masked
    nt_store(m,    &out[MASK_OFF + lin]);
}

// Exclusive scan of 16384 counts -> packed_info[ray] = {offset, count}.
// One 1024-thread block (32 wave32 waves); wave-shuffle scan + 32-entry LDS
// spine -> only 2 barriers instead of Hillis-Steele's 20.
__global__ __launch_bounds__(1024)
void scan_pack_kernel(const int* __restrict__ counts,
                      float*     __restrict__ out_packed)
{
    __shared__ int wsum[32];
    const int t    = threadIdx.x;
    const int lane = t & 31;
    const int wave = t >> 5;

    int local[16];
    int run = 0;
#pragma unroll
    for (int i = 0; i < 16; ++i) {
        local[i] = run;
        run += counts[t * 16 + i];
    }

    // inclusive wave32 scan of per-thread sums
    int incl = run;
#pragma unroll
    for (int off = 1; off < 32; off <<= 1) {
        int v = __shfl_up(incl, off, 32);
        if (lane >= off) incl += v;
    }
    if (lane == 31) wsum[wave] = incl;
    __syncthreads();

    // wave 0 scans the 32 wave totals into exclusive wave offsets
    if (wave == 0) {
        int v = wsum[lane];
        int wincl = v;
#pragma unroll
        for (int off = 1; off < 32; off <<= 1) {
            int u = __shfl_up(wincl, off, 32);
            if (lane >= off) wincl += u;
        }
        wsum[lane] = wincl - v;   // exclusive
    }
    __syncthreads();

    const int base = wsum[wave] + (incl - run);  // exclusive prefix for this thread

#pragma unroll
    for (int i = 0; i < 16; ++i) {
        const int ray = t * 16 + i;
        const int c   = counts[ray];
        // offsets/counts <= 2^23 -> exact in f32
        out_packed[2 * ray + 0] = (float)(base + local[i]);
        out_packed[2 * ray + 1] = (float)c;
    }
}

extern "C" void kernel_launch(void* const* d_in, const int* in_sizes, int n_in,
                              void* d_out, int out_size, void* d_ws, size_t ws_size,
                              hipStream_t stream)
{
    const float* rays_o = (const float*)d_in[0];
    const float* rays_d = (const float*)d_in[1];
    const float* aabb   = (const float*)d_in[2];
    const int*   occ    = (const int*)d_in[3];   // bool grid delivered as int32 per harness rule
    float* out    = (float*)d_out;
    int*   counts = (int*)d_ws;                  // 16384 * 4 = 64 KB scratch

    vol_sample_kernel<<<N_RAYS, MAX_STEPS, 0, stream>>>(rays_o, rays_d, aabb, occ,
                                                        out, counts);
    scan_pack_kernel<<<1, 1024, 0, stream>>>(counts, out + PACK_OFF);
}